// SPP_69853348102207
// MI455X (gfx1250) — compile-verified
//
#include <hip/hip_runtime.h>
#include <stdint.h>

// SPP: out = concat([x, maxpool5(x), maxpool9(x), maxpool13(x)], dim=C)
// x: [32, 512, 26, 26] f32.  Uses mp9 = mp5(mp5), mp13 = mp5(mp9) (exact with
// -inf SAME padding) and separable row/col 1D max passes, all inside LDS.
// Plane is staged into LDS with gfx1250 async-to-LDS loads (ASYNCcnt path),
// outputs streamed with non-temporal stores.

namespace {
constexpr int H       = 26;
constexpr int W       = 26;
constexpr int PLANE   = H * W;        // 676
constexpr int STRIDE  = 32;           // padded LDS row stride (words)
constexpr int ROWS    = H + 4;        // 2-row -inf halo top/bottom
constexpr int BUFW    = ROWS * STRIDE;// 960 words per buffer
constexpr int NTH     = 256;          // 8 wave32
constexpr int CIN     = 512;
constexpr int COUT    = 4 * CIN;
}

__global__ __launch_bounds__(NTH)
void spp_cascade_kernel(const float* __restrict__ x, float* __restrict__ out)
{
    __shared__ float A[BUFW];
    __shared__ float B[BUFW];

    const int tid   = threadIdx.x;
    const int plane = blockIdx.x;          // b*512 + c
    const int b     = plane >> 9;
    const int c     = plane & (CIN - 1);

    // ---- init both buffers (incl. halo) to -inf; interior is overwritten ----
    const float NEG_INF = -__builtin_inff();
    for (int i = tid; i < BUFW; i += NTH) { A[i] = NEG_INF; B[i] = NEG_INF; }
    __syncthreads();

    // ---- async DMA the 26x26 plane from global into LDS interior of A ----
    // Generic pointer to LDS: low 32 bits are the wave-relative LDS byte addr,
    // exactly what GLOBAL_LOAD_ASYNC_TO_LDS's VDST operand wants.
    const float*   src   = x + (size_t)plane * PLANE;
    const uint32_t a_lds = (uint32_t)(uintptr_t)(void*)&A[0];
    for (int i = tid; i < PLANE; i += NTH) {
        const int r  = i / W;
        const int cc = i - r * W;
        const uint32_t lds_addr =
            a_lds + (uint32_t)(((r + 2) * STRIDE + (cc + 2)) * sizeof(float));
        const float* g = src + i;
        asm volatile("global_load_async_to_lds_b32 %0, %1, off"
                     :: "v"(lds_addr), "v"(g)
                     : "memory");
    }
    asm volatile("s_wait_asynccnt 0" ::: "memory");
    __syncthreads();

    // ---- separable 5-wide max passes over the halo-padded LDS tiles ----
    auto row_pass = [&](const float* __restrict__ sb, float* __restrict__ db) {
        for (int i = tid; i < PLANE; i += NTH) {
            const int r  = i / W;
            const int cc = i - r * W;
            const int p  = (r + 2) * STRIDE + (cc + 2);
            float m = fmaxf(fmaxf(sb[p - 2], sb[p - 1]),
                            fmaxf(sb[p],
                                  fmaxf(sb[p + 1], sb[p + 2])));
            db[p] = m;
        }
    };
    auto col_pass = [&](const float* __restrict__ sb, float* __restrict__ db) {
        for (int i = tid; i < PLANE; i += NTH) {
            const int r  = i / W;
            const int cc = i - r * W;
            const int p  = (r + 2) * STRIDE + (cc + 2);
            float m = fmaxf(fmaxf(sb[p - 2 * STRIDE], sb[p - STRIDE]),
                            fmaxf(sb[p],
                                  fmaxf(sb[p + STRIDE], sb[p + 2 * STRIDE])));
            db[p] = m;
        }
    };
    // Streaming (write-once) output: non-temporal stores.
    auto store_sec = [&](const float* __restrict__ buf, int s) {
        float* o = out + ((size_t)b * COUT + (size_t)s * CIN + c) * PLANE;
        for (int i = tid; i < PLANE; i += NTH) {
            const int r  = i / W;
            const int cc = i - r * W;
            __builtin_nontemporal_store(buf[(r + 2) * STRIDE + (cc + 2)], o + i);
        }
    };

    store_sec(A, 0);                       // section 0: x itself

    #pragma unroll 1
    for (int s = 1; s <= 3; ++s) {         // A -> mp5 -> mp9 -> mp13
        row_pass(A, B);
        __syncthreads();
        col_pass(B, A);
        __syncthreads();
        store_sec(A, s);
    }
}

extern "C" void kernel_launch(void* const* d_in, const int* in_sizes, int n_in,
                              void* d_out, int out_size, void* d_ws, size_t ws_size,
                              hipStream_t stream)
{
    (void)in_sizes; (void)n_in; (void)out_size; (void)d_ws; (void)ws_size;
    const float* x   = (const float*)d_in[0];
    float*       out = (float*)d_out;

    const int nplanes = 32 * CIN;          // 16384 workgroups, one per (b,c)
    spp_cascade_kernel<<<dim3(nplanes), dim3(NTH), 0, stream>>>(x, out);
}